// HeteroGNN_46712064311418
// MI455X (gfx1250) — compile-verified
//
#include <hip/hip_runtime.h>

typedef float v2f __attribute__((ext_vector_type(2)));
typedef float v8f __attribute__((ext_vector_type(8)));

#define N_NODES 50000
#define F_IN    128
#define HID     64
#define N_EDGES 400000

// ---------------------------------------------------------------- utilities
__global__ void k_zero(float* __restrict__ p, long long n) {
  long long i = (long long)blockIdx.x * blockDim.x + threadIdx.x;
  if (i < n) p[i] = 0.0f;
}

// cnt[dst] += 1 per edge
__global__ void k_count(const int* __restrict__ ei, int E, float* __restrict__ cnt) {
  int e = blockIdx.x * blockDim.x + threadIdx.x;
  if (e < E) atomicAdd(&cnt[ei[E + e]], 1.0f);
}

// in-place: c = 1/max(c,1)
__global__ void k_rinv(float* __restrict__ c, int n) {
  int i = blockIdx.x * blockDim.x + threadIdx.x;
  if (i < n) c[i] = 1.0f / fmaxf(c[i], 1.0f);
}

// agg[dst,f] += x[src,f] for every edge
__global__ void k_scatter(const float* __restrict__ x, const int* __restrict__ ei,
                          int E, int F, float* __restrict__ agg) {
  long long gid = (long long)blockIdx.x * blockDim.x + threadIdx.x;
  if (gid >= (long long)E * F) return;
  int e = (int)(gid / F);
  int c = (int)(gid % F);
  int src = ei[e];
  int dst = ei[E + e];
  atomicAdd(&agg[(long long)dst * F + c], x[(long long)src * F + c]);
}

// Out[o,k] = sum_j WuA[o,col_off+j]*WsA[j,k]  (+ same for B pair if non-null)
// Wu: [64,128] row-major, Ws: [64,K] row-major, Out: [64,K]
__global__ void k_weight_prep(const float* __restrict__ WuA, const float* __restrict__ WsA,
                              const float* __restrict__ WuB, const float* __restrict__ WsB,
                              int col_off, int K, float* __restrict__ out) {
  int idx = blockIdx.x * blockDim.x + threadIdx.x;
  if (idx >= 64 * K) return;
  int o = idx / K, k = idx % K;
  float s = 0.0f;
  for (int j = 0; j < 64; ++j) s += WuA[o * 128 + col_off + j] * WsA[j * K + k];
  if (WuB)
    for (int j = 0; j < 64; ++j) s += WuB[o * 128 + col_off + j] * WsB[j * K + k];
  out[o * K + k] = s;
}

// ---------------------------------------------------------------- WMMA GEMM
// C[N,64] (+)= scale * ((diag(rowscale) * A[N,K]) @ W^T), W: [64,K] row-major.
// One wave -> 16 rows x 64 cols (4 WMMA C tiles), K stepped by 4 via
// V_WMMA_F32_16X16X4_F32 (exact fp32).
template <int K>
__global__ __launch_bounds__(128) void k_gemm_wmma(
    const float* __restrict__ A, const float* __restrict__ rowscale,
    const float* __restrict__ W, float* __restrict__ C,
    int N, float scale, int accumulate) {
  const int wave = threadIdx.x >> 5;
  const int lane = threadIdx.x & 31;
  const int half = lane >> 4;   // 0: K=0,1 ; 1: K=2,3 within a k-step
  const int l16  = lane & 15;
  const int r0 = ((blockIdx.x << 2) + wave) << 4;   // 16-row strip base
  if (r0 >= N) return;          // wave-uniform: EXEC stays all-ones for WMMA

  const int arow = r0 + l16;
  const float rs = rowscale ? rowscale[arow] : 1.0f;
  const float* __restrict__ Ap = A + (long long)arow * K + 2 * half;
  const float* __restrict__ Wp = W + l16 * K + 2 * half;

  v8f acc0 = {}, acc1 = {}, acc2 = {}, acc3 = {};
#pragma unroll 8
  for (int k0 = 0; k0 < K; k0 += 4) {
    v2f a = *(const v2f*)(Ap + k0);
    a *= rs;  // fold mean-divisor into A
    v2f b0 = *(const v2f*)(Wp + 0 * 16 * K + k0);
    v2f b1 = *(const v2f*)(Wp + 1 * 16 * K + k0);
    v2f b2 = *(const v2f*)(Wp + 2 * 16 * K + k0);
    v2f b3 = *(const v2f*)(Wp + 3 * 16 * K + k0);
    acc0 = __builtin_amdgcn_wmma_f32_16x16x4_f32(false, a, false, b0, (short)0, acc0, false, false);
    acc1 = __builtin_amdgcn_wmma_f32_16x16x4_f32(false, a, false, b1, (short)0, acc1, false, false);
    acc2 = __builtin_amdgcn_wmma_f32_16x16x4_f32(false, a, false, b2, (short)0, acc2, false, false);
    acc3 = __builtin_amdgcn_wmma_f32_16x16x4_f32(false, a, false, b3, (short)0, acc3, false, false);
  }

  const v8f accs[4] = {acc0, acc1, acc2, acc3};
#pragma unroll
  for (int c = 0; c < 4; ++c) {
#pragma unroll
    for (int v = 0; v < 8; ++v) {
      // C layout: VGPR v, lanes 0-15 -> M=v, lanes 16-31 -> M=8+v; N = l16
      long long idx = (long long)(r0 + v + 8 * half) * 64 + c * 16 + l16;
      float val = scale * accs[c][v];
      if (accumulate) val += C[idx];
      C[idx] = val;
    }
  }
}

// ---------------------------------------------------------------- BatchNorm
// acc[0..63] += sum(h[:,f]); acc[64..127] += sum(h[:,f]^2)
__global__ void k_bn_stats(const float* __restrict__ h, int N, float* __restrict__ acc) {
  __shared__ float ls[128];
  if (threadIdx.x < 128) ls[threadIdx.x] = 0.0f;
  __syncthreads();
  const long long total  = (long long)N * 64;
  const long long stride = (long long)gridDim.x * blockDim.x;  // multiple of 64
  long long i = (long long)blockIdx.x * blockDim.x + threadIdx.x;
  float s = 0.0f, sq = 0.0f;
  for (; i < total; i += stride) {
    float v = h[i];
    s += v;
    sq += v * v;
  }
  int f = threadIdx.x & 63;  // feature is invariant across the strided loop
  atomicAdd(&ls[f], s);
  atomicAdd(&ls[64 + f], sq);
  __syncthreads();
  if (threadIdx.x < 128) atomicAdd(&acc[threadIdx.x], ls[threadIdx.x]);
}

// coef[f] = g*rsqrt(var+1); coef[64+f] = b - mean*coef[f]   (eps = 1.0, biased var)
__global__ void k_bn_finalize(const float* __restrict__ acc, const float* __restrict__ g,
                              const float* __restrict__ b, float invN,
                              float* __restrict__ coef) {
  int f = threadIdx.x;
  if (f < 64) {
    float mean = acc[f] * invN;
    float var  = acc[64 + f] * invN - mean * mean;
    float sc   = g[f] * rsqrtf(var + 1.0f);
    coef[f]      = sc;
    coef[64 + f] = b[f] - mean * sc;
  }
}

// x = leaky_relu(h*coef_scale + coef_shift, 0.01)
__global__ void k_bn_apply(const float* __restrict__ h, const float* __restrict__ coef,
                           long long n, float* __restrict__ x) {
  long long i = (long long)blockIdx.x * blockDim.x + threadIdx.x;
  if (i >= n) return;
  int f = (int)(i & 63);
  float v = h[i] * coef[f] + coef[64 + f];
  x[i] = v > 0.0f ? v : 0.01f * v;
}

// ---------------------------------------------------------------- edge gather
// out[row, :] = [x[src] || x[dst]] ; rows 0..E-1 from ei0, E..2E-1 from ei1
__global__ void k_edge_gather(const float* __restrict__ x, const int* __restrict__ ei0,
                              const int* __restrict__ ei1, int E, float4* __restrict__ out) {
  long long gid = (long long)blockIdx.x * blockDim.x + threadIdx.x;
  long long total = (long long)2 * E * 32;  // 32 float4 per 128-float row
  if (gid >= total) return;
  int c4 = (int)(gid & 31);
  long long row = gid >> 5;
  const int* ei = (row < E) ? ei0 : ei1;
  long long e   = (row < E) ? row : row - E;
  int node = (c4 < 16) ? ei[e] : ei[E + e];
  int cc   = (c4 < 16) ? c4 : c4 - 16;
  out[row * 32 + c4] = ((const float4*)x)[(long long)node * 16 + cc];
}

// ---------------------------------------------------------------- host side
static inline int cdiv(long long a, int b) { return (int)((a + b - 1) / b); }

static void run_layer(const float* xin, int K,
                      const float* Wd0, const float* Ws0, const float* Wu0,
                      const float* Wd1, const float* Ws1, const float* Wu1,
                      const float* g, const float* b,
                      const int* ei0, const int* ei1,
                      const float* rinv0, const float* rinv1,
                      float* agg, float* h, float* Wbuf, float* bnacc, float* coef,
                      float* xout, hipStream_t stream) {
  const int N = N_NODES, E = N_EDGES;
  const int gemmBlocks = cdiv(N / 16, 4);      // 4 waves/block, 16 rows/wave
  const long long aggN = (long long)N * K;
  const long long scN  = (long long)E * K;
  const long long hN   = (long long)N * 64;

  // dst path for both edge types fused: Wbuf = WuL0*Wd0 + WuL1*Wd1, h = 0.5*x@Wbuf^T
  k_weight_prep<<<cdiv(64 * K, 256), 256, 0, stream>>>(Wu0, Wd0, Wu1, Wd1, 0, K, Wbuf);
  if (K == 128) k_gemm_wmma<128><<<gemmBlocks, 128, 0, stream>>>(xin, nullptr, Wbuf, h, N, 0.5f, 0);
  else          k_gemm_wmma<64><<<gemmBlocks, 128, 0, stream>>>(xin, nullptr, Wbuf, h, N, 0.5f, 0);

  // edge type 0: h += 0.5 * mean_agg0(x) @ (WuR0*Ws0)^T
  k_zero<<<cdiv(aggN, 256), 256, 0, stream>>>(agg, aggN);
  k_scatter<<<cdiv(scN, 256), 256, 0, stream>>>(xin, ei0, E, K, agg);
  k_weight_prep<<<cdiv(64 * K, 256), 256, 0, stream>>>(Wu0, Ws0, nullptr, nullptr, 64, K, Wbuf);
  if (K == 128) k_gemm_wmma<128><<<gemmBlocks, 128, 0, stream>>>(agg, rinv0, Wbuf, h, N, 0.5f, 1);
  else          k_gemm_wmma<64><<<gemmBlocks, 128, 0, stream>>>(agg, rinv0, Wbuf, h, N, 0.5f, 1);

  // edge type 1: h += 0.5 * mean_agg1(x) @ (WuR1*Ws1)^T
  k_zero<<<cdiv(aggN, 256), 256, 0, stream>>>(agg, aggN);
  k_scatter<<<cdiv(scN, 256), 256, 0, stream>>>(xin, ei1, E, K, agg);
  k_weight_prep<<<cdiv(64 * K, 256), 256, 0, stream>>>(Wu1, Ws1, nullptr, nullptr, 64, K, Wbuf);
  if (K == 128) k_gemm_wmma<128><<<gemmBlocks, 128, 0, stream>>>(agg, rinv1, Wbuf, h, N, 0.5f, 1);
  else          k_gemm_wmma<64><<<gemmBlocks, 128, 0, stream>>>(agg, rinv1, Wbuf, h, N, 0.5f, 1);

  // BatchNorm (training mode, eps=1) + LeakyReLU(0.01)
  k_zero<<<1, 128, 0, stream>>>(bnacc, 128);
  k_bn_stats<<<512, 256, 0, stream>>>(h, N, bnacc);
  k_bn_finalize<<<1, 64, 0, stream>>>(bnacc, g, b, 1.0f / (float)N, coef);
  k_bn_apply<<<cdiv(hN, 256), 256, 0, stream>>>(h, coef, hN, xout);
}

extern "C" void kernel_launch(void* const* d_in, const int* in_sizes, int n_in,
                              void* d_out, int out_size, void* d_ws, size_t ws_size,
                              hipStream_t stream) {
  const int N = N_NODES, E = N_EDGES;
  const float* x0  = (const float*)d_in[0];
  const int*   ei0 = (const int*)d_in[1];
  const int*   ei1 = (const int*)d_in[2];
  const float* W[12];
  for (int i = 0; i < 12; ++i) W[i] = (const float*)d_in[3 + i];
  // W[0..5]  = l1t0_Wd, l1t0_Ws, l1t0_Wu, l1t1_Wd, l1t1_Ws, l1t1_Wu
  // W[6..11] = l2t0_Wd, l2t0_Ws, l2t0_Wu, l2t1_Wd, l2t1_Ws, l2t1_Wu
  const float* bn1g = (const float*)d_in[15];
  const float* bn1b = (const float*)d_in[16];
  const float* bn2g = (const float*)d_in[17];
  const float* bn2b = (const float*)d_in[18];

  float* ws = (float*)d_ws;
  float* rinv0 = ws; ws += N;
  float* rinv1 = ws; ws += N;
  float* agg   = ws; ws += (long long)N * 128;
  float* h     = ws; ws += (long long)N * 64;
  float* x1    = ws; ws += (long long)N * 64;
  float* x2    = ws; ws += (long long)N * 64;
  float* Wbuf  = ws; ws += 64 * 128;
  float* bnacc = ws; ws += 128;
  float* coef  = ws; ws += 128;

  // per-destination edge counts -> reciprocals (shared by both layers)
  k_zero<<<cdiv(N, 256), 256, 0, stream>>>(rinv0, N);
  k_zero<<<cdiv(N, 256), 256, 0, stream>>>(rinv1, N);
  k_count<<<cdiv(E, 256), 256, 0, stream>>>(ei0, E, rinv0);
  k_count<<<cdiv(E, 256), 256, 0, stream>>>(ei1, E, rinv1);
  k_rinv<<<cdiv(N, 256), 256, 0, stream>>>(rinv0, N);
  k_rinv<<<cdiv(N, 256), 256, 0, stream>>>(rinv1, N);

  // layer 1 (F_IN=128) and layer 2 (HID=64)
  run_layer(x0, 128, W[0], W[1], W[2], W[3], W[4], W[5], bn1g, bn1b,
            ei0, ei1, rinv0, rinv1, agg, h, Wbuf, bnacc, coef, x1, stream);
  run_layer(x1, 64, W[6], W[7], W[8], W[9], W[10], W[11], bn2g, bn2b,
            ei0, ei1, rinv0, rinv1, agg, h, Wbuf, bnacc, coef, x2, stream);

  // edge embeddings: [x[src] || x[dst]] per edge, e0 then e1 -> [2E, 128]
  long long gatherN = (long long)2 * E * 32;
  k_edge_gather<<<cdiv(gatherN, 256), 256, 0, stream>>>(x2, ei0, ei1, E, (float4*)d_out);
}